// STVimTokenMambaEncoder_17781164605845
// MI455X (gfx1250) — compile-verified
//
#include <hip/hip_runtime.h>
#include <math.h>

// ---------------------------------------------------------------------------
// STVimTokenMambaEncoder forward for MI455X (gfx1250, wave32, WMMA).
// Latency-bound workload (~7 GFLOP, working set << 192MB L2): all GEMMs use
// V_WMMA_F32_16X16X4_F32 (fp32 exact vs. reference), everything else VALU.
// Each GEMM wave computes a 16x32 output tile (A fragment reused across two
// B tiles) with an unscaled fast path when the column pair is fully in range.
// ---------------------------------------------------------------------------

#define BSZ      4
#define TT       8
#define NSEQ_V   32          // B*T frames
#define NPATCH   64
#define LV       65          // NPATCH + cls
#define CLS_POS  32
#define EMBED    192
#define DINNER   384
#define DSTATE   16
#define DTRANK   12
#define XPN      44          // DTRANK + 2*DSTATE
#define PATCHK   256         // 16*16*1
#define NTOK_P   2048        // 32 frames * 64 patches

typedef __attribute__((ext_vector_type(2))) float v2f;
typedef __attribute__((ext_vector_type(8))) float v8f;

__device__ __forceinline__ float silu_(float x) { return x / (1.0f + __expf(-x)); }
__device__ __forceinline__ float softplus_(float x) {
  return x > 20.0f ? x : log1pf(__expf(x));
}

// ------------------------- generic WMMA f32 GEMM ---------------------------
// C[M x Nact] = act(A[M x K] * W[Nact x K]^T + bias),  M mult of 16, K of 4.
// One wave computes a 16x32 tile (two 16x16 WMMA accumulators sharing the A
// fragment). EXEC stays all-ones through the WMMA loops (tails handled by
// zero-scaling B, not divergence), per ISA 7.12 restrictions.
__global__ void wgemm_f32_kernel(const float* __restrict__ A, int lda,
                                 const float* __restrict__ W, int ldw,
                                 const float* __restrict__ bias,
                                 float* __restrict__ C, int ldc,
                                 int Mrows, int K, int Nact, int act) {
  const int lane   = threadIdx.x;               // 0..31
  const int mtile  = Mrows >> 4;
  const int ntile16 = (Nact + 15) >> 4;
  const int npair  = (ntile16 + 1) >> 1;
  const int tile   = blockIdx.x * blockDim.y + threadIdx.y;
  if (tile >= mtile * npair) return;            // wave-uniform
  const int tm  = (tile % mtile) << 4;
  const int tp  = tile / mtile;
  const int tn0 = tp << 5;                      // first 16-col tile
  const bool two = (2 * tp + 1) < ntile16;      // second 16-col tile exists

  const int l15  = lane & 15;
  const int half = lane >> 4;                   // 0: K=0,1  1: K=2,3
  const float* arow = A + (size_t)(tm + l15) * lda;
  const int col0 = tn0 + l15;
  const int col1 = col0 + 16;

  v8f acc0 = {}, acc1 = {};

  if (two && (tn0 + 32) <= Nact) {
    // ---- fast path: both column tiles fully in range, no scaling ----
    const float* w0 = W + (size_t)col0 * ldw;
    const float* w1 = W + (size_t)col1 * ldw;
    for (int k0 = 0; k0 < K; k0 += 4) {
      const int kb = k0 + half * 2;
      v2f a, b0, b1;
      a[0]  = arow[kb]; a[1]  = arow[kb + 1];
      b0[0] = w0[kb];   b0[1] = w0[kb + 1];
      b1[0] = w1[kb];   b1[1] = w1[kb + 1];
      acc0 = __builtin_amdgcn_wmma_f32_16x16x4_f32(
          false, a, false, b0, (short)0, acc0, false, false);
      acc1 = __builtin_amdgcn_wmma_f32_16x16x4_f32(
          false, a, false, b1, (short)0, acc1, false, false);
    }
  } else {
    // ---- guarded path (N tail): zero-scale out-of-range columns ----
    const bool nok0 = col0 < Nact;
    const bool nok1 = two && (col1 < Nact);
    const float s0 = nok0 ? 1.0f : 0.0f;
    const float s1 = nok1 ? 1.0f : 0.0f;
    const float* w0 = W + (size_t)(nok0 ? col0 : 0) * ldw;
    const float* w1 = W + (size_t)(nok1 ? col1 : 0) * ldw;
    if (two) {
      for (int k0 = 0; k0 < K; k0 += 4) {
        const int kb = k0 + half * 2;
        v2f a, b0, b1;
        a[0]  = arow[kb];       a[1]  = arow[kb + 1];
        b0[0] = w0[kb] * s0;    b0[1] = w0[kb + 1] * s0;
        b1[0] = w1[kb] * s1;    b1[1] = w1[kb + 1] * s1;
        acc0 = __builtin_amdgcn_wmma_f32_16x16x4_f32(
            false, a, false, b0, (short)0, acc0, false, false);
        acc1 = __builtin_amdgcn_wmma_f32_16x16x4_f32(
            false, a, false, b1, (short)0, acc1, false, false);
      }
    } else {
      for (int k0 = 0; k0 < K; k0 += 4) {
        const int kb = k0 + half * 2;
        v2f a, b0;
        a[0]  = arow[kb];       a[1]  = arow[kb + 1];
        b0[0] = w0[kb] * s0;    b0[1] = w0[kb + 1] * s0;
        acc0 = __builtin_amdgcn_wmma_f32_16x16x4_f32(
            false, a, false, b0, (short)0, acc0, false, false);
      }
    }
  }

  // ---- store (divergence OK after WMMAs) ----
  if (col0 < Nact) {
    const float bv = bias ? bias[col0] : 0.0f;
#pragma unroll
    for (int r = 0; r < 8; ++r) {
      const int row = tm + r + half * 8;        // C/D layout: VGPR r -> M=r(+8)
      float v = acc0[r] + bv;
      if (act == 1) v = softplus_(v);
      C[(size_t)row * ldc + col0] = v;
    }
  }
  if (two && col1 < Nact) {
    const float bv = bias ? bias[col1] : 0.0f;
#pragma unroll
    for (int r = 0; r < 8; ++r) {
      const int row = tm + r + half * 8;
      float v = acc1[r] + bv;
      if (act == 1) v = softplus_(v);
      C[(size_t)row * ldc + col1] = v;
    }
  }
}

// --------------------------- data-movement kernels -------------------------
__global__ void patchify_kernel(const float* __restrict__ depth,
                                float* __restrict__ patches) {
  int idx = blockIdx.x * blockDim.x + threadIdx.x;
  if (idx >= NTOK_P * PATCHK) return;
  int k   = idx % PATCHK;
  int tok = idx / PATCHK;
  int f = tok / NPATCH, p = tok % NPATCH;
  int ph = p >> 3, pw = p & 7;
  int r = k >> 4, c = k & 15;
  patches[idx] = depth[(size_t)f * 128 * 128 + (ph * 16 + r) * 128 + (pw * 16 + c)];
}

// x0[m][l][e] = (l==CLS ? cls : tokemb) + pos_embed
__global__ void assemble_kernel(const float* __restrict__ tokemb,
                                const float* __restrict__ cls_token,
                                const float* __restrict__ pos_embed,
                                float* __restrict__ x0) {
  int idx = blockIdx.x * blockDim.x + threadIdx.x;
  if (idx >= NSEQ_V * LV * EMBED) return;
  int e = idx % EMBED;
  int rest = idx / EMBED;
  int l = rest % LV, m = rest / LV;
  float v;
  if (l == CLS_POS) v = cls_token[e];
  else {
    int lp = (l < CLS_POS) ? l : l - 1;
    v = tokemb[((size_t)m * NPATCH + lp) * EMBED + e];
  }
  x0[idx] = v + pos_embed[l * EMBED + e];
}

// residual = hid (+ res); normed = LN(residual)*w+b.  One wave per token.
__global__ void ln_resid_kernel(const float* __restrict__ hid,
                                float* __restrict__ res,
                                const float* __restrict__ w,
                                const float* __restrict__ b,
                                float* __restrict__ out, int first) {
  const int tok = blockIdx.x;
  const int lane = threadIdx.x;                 // 32
  const float* hrow = hid + (size_t)tok * EMBED;
  float* rrow = res + (size_t)tok * EMBED;
  float vals[6];
  float s = 0.0f;
#pragma unroll
  for (int j = 0; j < 6; ++j) {
    int e = lane + 32 * j;
    float v = hrow[e] + (first ? 0.0f : rrow[e]);
    vals[j] = v; s += v;
  }
  for (int off = 16; off; off >>= 1) s += __shfl_xor(s, off, 32);
  float mean = s * (1.0f / EMBED);
  float vs = 0.0f;
#pragma unroll
  for (int j = 0; j < 6; ++j) { float d = vals[j] - mean; vs += d * d; }
  for (int off = 16; off; off >>= 1) vs += __shfl_xor(vs, off, 32);
  float rstd = rsqrtf(vs * (1.0f / EMBED) + 1e-5f);
  float* orow = out + (size_t)tok * EMBED;
#pragma unroll
  for (int j = 0; j < 6; ++j) {
    int e = lane + 32 * j;
    rrow[e] = vals[j];
    orow[e] = (vals[j] - mean) * rstd * w[e] + b[e];
  }
}

// causal depthwise conv (K=4) + SiLU; dir=1 runs on the time-reversed sequence.
__global__ void conv_silu_kernel(const float* __restrict__ xz,
                                 const float* __restrict__ cw,
                                 const float* __restrict__ cb,
                                 float* __restrict__ xc,
                                 int nseq, int L, int dir) {
  int idx = blockIdx.x * blockDim.x + threadIdx.x;
  if (idx >= nseq * L * DINNER) return;
  int d = idx % DINNER;
  int rest = idx / DINNER;
  int i = rest % L, m = rest / L;
  float acc = cb[d];
#pragma unroll
  for (int t = 0; t < 4; ++t) {
    int j = i - 3 + t;
    if (j < 0) continue;
    int l = dir ? (L - 1 - j) : j;
    acc += cw[d * 4 + t] * xz[((size_t)m * L + l) * (2 * DINNER) + d];
  }
  xc[idx] = silu_(acc);
}

// selective scan: one thread per (seq, channel), 16-state recurrence in regs.
// Applies SiLU(z) gating and accumulates into y_total at the original time.
__global__ void scan_kernel(const float* __restrict__ xc,
                            const float* __restrict__ dt,
                            const float* __restrict__ proj,
                            const float* __restrict__ xz,
                            const float* __restrict__ A_log,
                            const float* __restrict__ Dp,
                            float* __restrict__ ytot,
                            int nseq, int L, int dir) {
  int idx = blockIdx.x * blockDim.x + threadIdx.x;
  if (idx >= nseq * DINNER) return;
  int d = idx % DINNER, m = idx / DINNER;
  float negA[DSTATE], h[DSTATE];
#pragma unroll
  for (int n = 0; n < DSTATE; ++n) { negA[n] = -__expf(A_log[d * DSTATE + n]); h[n] = 0.0f; }
  const float Dv = Dp[d];
  for (int i = 0; i < L; ++i) {
    size_t tok = (size_t)m * L + i;
    float u   = xc[tok * DINNER + d];
    float dtv = dt[tok * DINNER + d];
    const float* pr = proj + tok * XPN;
    float du = dtv * u;
    float y = 0.0f;
#pragma unroll
    for (int n = 0; n < DSTATE; ++n) {
      float dA = __expf(dtv * negA[n]);
      h[n] = dA * h[n] + du * pr[DTRANK + n];
      y += h[n] * pr[DTRANK + DSTATE + n];
    }
    y += u * Dv;
    int l = dir ? (L - 1 - i) : i;
    float z = xz[((size_t)m * L + l) * (2 * DINNER) + DINNER + d];
    ytot[((size_t)m * L + l) * DINNER + d] += y * silu_(z);
  }
}

__global__ void zero_kernel(float* __restrict__ p, int n) {
  int i = blockIdx.x * blockDim.x + threadIdx.x;
  if (i < n) p[i] = 0.0f;
}

// feat[m][:] = LN(hid+res)[token m*65+CLS] -> ht (== (B,T,E) row-major)
__global__ void final_feat_kernel(const float* __restrict__ hid,
                                  const float* __restrict__ res,
                                  const float* __restrict__ w,
                                  const float* __restrict__ b,
                                  float* __restrict__ ht) {
  const int m = blockIdx.x;
  const int lane = threadIdx.x;
  const size_t tok = (size_t)m * LV + CLS_POS;
  float vals[6];
  float s = 0.0f;
#pragma unroll
  for (int j = 0; j < 6; ++j) {
    int e = lane + 32 * j;
    float v = hid[tok * EMBED + e] + res[tok * EMBED + e];
    vals[j] = v; s += v;
  }
  for (int off = 16; off; off >>= 1) s += __shfl_xor(s, off, 32);
  float mean = s * (1.0f / EMBED);
  float vs = 0.0f;
#pragma unroll
  for (int j = 0; j < 6; ++j) { float d = vals[j] - mean; vs += d * d; }
  for (int off = 16; off; off >>= 1) vs += __shfl_xor(vs, off, 32);
  float rstd = rsqrtf(vs * (1.0f / EMBED) + 1e-5f);
#pragma unroll
  for (int j = 0; j < 6; ++j) {
    int e = lane + 32 * j;
    ht[(size_t)m * EMBED + e] = (vals[j] - mean) * rstd * w[e] + b[e];
  }
}

// out[b][e] = ht[b, T-1, e] + MLP(concat(state,action))
__global__ void head_kernel(const float* __restrict__ ht,
                            const float* __restrict__ sv,
                            const float* __restrict__ ac,
                            const float* __restrict__ w1, const float* __restrict__ b1,
                            const float* __restrict__ w2, const float* __restrict__ b2,
                            float* __restrict__ out) {
  __shared__ float cin[40];
  __shared__ float h1[EMBED];
  const int bb = blockIdx.x;
  const int t = threadIdx.x;                    // 192
  if (t < 32) cin[t] = sv[bb * 32 + t];
  else if (t < 40) cin[t] = ac[bb * 8 + (t - 32)];
  __syncthreads();
  float a1 = b1[t];
  for (int k = 0; k < 40; ++k) a1 += w1[t * 40 + k] * cin[k];
  h1[t] = fmaxf(a1, 0.0f);
  __syncthreads();
  float o = b2[t];
  for (int j = 0; j < EMBED; ++j) o += w2[t * EMBED + j] * h1[j];
  out[bb * EMBED + t] = o + ht[((size_t)bb * TT + (TT - 1)) * EMBED + t];
}

// ------------------------------ host side ----------------------------------
static inline void launch_gemm(const float* A, int lda, const float* W, int ldw,
                               const float* bias, float* C, int ldc,
                               int M, int K, int Nact, int act, hipStream_t s) {
  int ntile16 = (Nact + 15) / 16;
  int npair   = (ntile16 + 1) / 2;
  int tiles   = (M / 16) * npair;
  dim3 blk(32, 4);
  int grid = (tiles + 3) / 4;
  wgemm_f32_kernel<<<grid, blk, 0, s>>>(A, lda, W, ldw, bias, C, ldc, M, K, Nact, act);
}

extern "C" void kernel_launch(void* const* d_in, const int* in_sizes, int n_in,
                              void* d_out, int out_size, void* d_ws, size_t ws_size,
                              hipStream_t stream) {
  (void)in_sizes; (void)n_in; (void)out_size; (void)ws_size;
  // pytree-flattened inputs (dicts alphabetical):
  const float* depth     = (const float*)d_in[0];
  const float* state_vec = (const float*)d_in[1];
  const float* action    = (const float*)d_in[2];
  const float* cls_token = (const float*)d_in[3];
  const float* norm_f_b  = (const float*)d_in[4];
  const float* norm_f_w  = (const float*)d_in[5];
  const float* patch_b   = (const float*)d_in[6];
  const float* patch_w   = (const float*)d_in[7];
  const float* pos_embed = (const float*)d_in[8];
  const float* vA_log    = (const float*)d_in[9];
  const float* vA_log_b  = (const float*)d_in[10];
  const float* vDp       = (const float*)d_in[11];
  const float* vDp_b     = (const float*)d_in[12];
  const float* vconv_b   = (const float*)d_in[13];
  const float* vconv_b_b = (const float*)d_in[14];
  const float* vconv_w   = (const float*)d_in[15];
  const float* vconv_w_b = (const float*)d_in[16];
  const float* vdt_b     = (const float*)d_in[17];
  const float* vdt_b_b   = (const float*)d_in[18];
  const float* vdt_w     = (const float*)d_in[19];
  const float* vdt_w_b   = (const float*)d_in[20];
  const float* vin_proj  = (const float*)d_in[21];
  const float* vln_b     = (const float*)d_in[22];
  const float* vln_w     = (const float*)d_in[23];
  const float* vout_proj = (const float*)d_in[24];
  const float* vxproj    = (const float*)d_in[25];
  const float* vxproj_b  = (const float*)d_in[26];
  const float* tA_log    = (const float*)d_in[27];
  const float* tDp       = (const float*)d_in[28];
  const float* tconv_b   = (const float*)d_in[29];
  const float* tconv_w   = (const float*)d_in[30];
  const float* tdt_b     = (const float*)d_in[31];
  const float* tdt_w     = (const float*)d_in[32];
  const float* tin_proj  = (const float*)d_in[33];
  const float* tout_proj = (const float*)d_in[34];
  const float* txproj    = (const float*)d_in[35];
  const float* cb1       = (const float*)d_in[36];
  const float* cb2       = (const float*)d_in[37];
  const float* cw1       = (const float*)d_in[38];
  const float* cw2       = (const float*)d_in[39];

  float* ws = (float*)d_ws;
  // workspace layout (floats), ~25 MB total
  float* patches = ws;                                  size_t off = (size_t)NTOK_P * PATCHK;
  float* tokemb  = ws + off;                            off += (size_t)NTOK_P * EMBED;
  float* hid     = ws + off;                            off += (size_t)NSEQ_V * LV * EMBED;
  float* res     = ws + off;                            off += (size_t)NSEQ_V * LV * EMBED;
  float* normed  = ws + off;                            off += (size_t)NSEQ_V * LV * EMBED;
  float* xz      = ws + off;                            off += (size_t)NSEQ_V * LV * 2 * DINNER;
  float* xc      = ws + off;                            off += (size_t)NSEQ_V * LV * DINNER;
  float* projb   = ws + off;                            off += (size_t)NSEQ_V * LV * XPN;
  float* dtb     = ws + off;                            off += (size_t)NSEQ_V * LV * DINNER;
  float* ytot    = ws + off;                            off += (size_t)NSEQ_V * LV * DINNER;
  float* ht      = ws + off;

  const int TOKV = NSEQ_V * LV;                          // 2080
  // ---- patch embed ----
  patchify_kernel<<<(NTOK_P * PATCHK + 255) / 256, 256, 0, stream>>>(depth, patches);
  launch_gemm(patches, PATCHK, patch_w, PATCHK, patch_b, tokemb, EMBED,
              NTOK_P, PATCHK, EMBED, 0, stream);
  assemble_kernel<<<(TOKV * EMBED + 255) / 256, 256, 0, stream>>>(tokemb, cls_token, pos_embed, hid);

  // ---- 8 bidirectional Vim layers ----
  for (int i = 0; i < 8; ++i) {
    ln_resid_kernel<<<TOKV, 32, 0, stream>>>(hid, res, vln_w + i * EMBED, vln_b + i * EMBED,
                                             normed, i == 0);
    launch_gemm(normed, EMBED, vin_proj + (size_t)i * 2 * DINNER * EMBED, EMBED,
                nullptr, xz, 2 * DINNER, TOKV, EMBED, 2 * DINNER, 0, stream);
    zero_kernel<<<(TOKV * DINNER + 255) / 256, 256, 0, stream>>>(ytot, TOKV * DINNER);
    for (int dir = 0; dir < 2; ++dir) {
      const float* cw = (dir ? vconv_w_b : vconv_w) + (size_t)i * DINNER * 4;
      const float* cb = (dir ? vconv_b_b : vconv_b) + (size_t)i * DINNER;
      const float* xp = (dir ? vxproj_b : vxproj)   + (size_t)i * XPN * DINNER;
      const float* dw = (dir ? vdt_w_b : vdt_w)     + (size_t)i * DINNER * DTRANK;
      const float* db = (dir ? vdt_b_b : vdt_b)     + (size_t)i * DINNER;
      const float* al = (dir ? vA_log_b : vA_log)   + (size_t)i * DINNER * DSTATE;
      const float* dp = (dir ? vDp_b : vDp)         + (size_t)i * DINNER;
      conv_silu_kernel<<<(TOKV * DINNER + 255) / 256, 256, 0, stream>>>(xz, cw, cb, xc,
                                                                        NSEQ_V, LV, dir);
      launch_gemm(xc, DINNER, xp, DINNER, nullptr, projb, XPN, TOKV, DINNER, XPN, 0, stream);
      launch_gemm(projb, XPN, dw, DTRANK, db, dtb, DINNER, TOKV, DTRANK, DINNER, 1, stream);
      scan_kernel<<<(NSEQ_V * DINNER + 255) / 256, 256, 0, stream>>>(
          xc, dtb, projb, xz, al, dp, ytot, NSEQ_V, LV, dir);
    }
    launch_gemm(ytot, DINNER, vout_proj + (size_t)i * EMBED * DINNER, DINNER,
                nullptr, hid, EMBED, TOKV, DINNER, EMBED, 0, stream);
  }
  final_feat_kernel<<<NSEQ_V, 32, 0, stream>>>(hid, res, norm_f_w, norm_f_b, ht);

  // ---- 2 temporal (unidirectional) Mamba layers on (B=4, T=8) ----
  const int TOKT = BSZ * TT;                             // 32
  for (int i = 0; i < 2; ++i) {
    launch_gemm(ht, EMBED, tin_proj + (size_t)i * 2 * DINNER * EMBED, EMBED,
                nullptr, xz, 2 * DINNER, TOKT, EMBED, 2 * DINNER, 0, stream);
    zero_kernel<<<(TOKT * DINNER + 255) / 256, 256, 0, stream>>>(ytot, TOKT * DINNER);
    conv_silu_kernel<<<(TOKT * DINNER + 255) / 256, 256, 0, stream>>>(
        xz, tconv_w + (size_t)i * DINNER * 4, tconv_b + (size_t)i * DINNER, xc, BSZ, TT, 0);
    launch_gemm(xc, DINNER, txproj + (size_t)i * XPN * DINNER, DINNER,
                nullptr, projb, XPN, TOKT, DINNER, XPN, 0, stream);
    launch_gemm(projb, XPN, tdt_w + (size_t)i * DINNER * DTRANK, DTRANK,
                tdt_b + (size_t)i * DINNER, dtb, DINNER, TOKT, DTRANK, DINNER, 1, stream);
    scan_kernel<<<(BSZ * DINNER + 255) / 256, 256, 0, stream>>>(
        xc, dtb, projb, xz, tA_log + (size_t)i * DINNER * DSTATE,
        tDp + (size_t)i * DINNER, ytot, BSZ, TT, 0);
    launch_gemm(ytot, DINNER, tout_proj + (size_t)i * EMBED * DINNER, DINNER,
                nullptr, ht, EMBED, TOKT, DINNER, EMBED, 0, stream);
  }

  head_kernel<<<BSZ, EMBED, 0, stream>>>(ht, state_vec, action, cw1, cb1, cw2, cb2,
                                         (float*)d_out);
}